// RegressionLoss_60181081752185
// MI455X (gfx1250) — compile-verified
//
#include <hip/hip_runtime.h>
#include <stdint.h>

// ---------------------------------------------------------------------------
// ATSS 1D regression loss, MI455X / gfx1250.
// Sparse-candidate formulation: only 135 candidates per (batch, gt) can be
// positive, so we never build the A x G matrices. Kernel A finds exact
// top-27-per-level candidates (uniform-grid window + exact float rank with
// index tie-break), computes IoUs, and uses V_WMMA_F32_16X16X4_F32 with a
// ones B-matrix to produce per-gt sum / sum-of-squared-deviation reductions
// in full fp32. Kernel B resolves per-anchor argmax over gts (same-level
// candidate scan) and accumulates the masked L1 loss with a deterministic
// tree reduction. Kernel C averages the 4 per-batch losses.
// ---------------------------------------------------------------------------

#define B_SZ 4
#define G_SZ 48
#define NLEV 5
#define TOPK 27
#define CPG  (NLEV * TOPK)        // 135 candidates per gt
#define RECS (G_SZ * CPG)         // 6480 candidate records per batch item
#define PITCH 136                 // candidate row padded with one zero column
#define A_TOTAL 380928

typedef __attribute__((ext_vector_type(2))) float v2f;
typedef __attribute__((ext_vector_type(8))) float v8f;

__constant__ int d_stride[NLEV] = {2, 4, 8, 16, 32};
__constant__ int d_npos[NLEV]   = {65536, 32768, 16384, 8192, 4096};
__constant__ int d_aoff[NLEV]   = {0, 196608, 294912, 344064, 368640};

#if defined(__has_builtin)
#if __has_builtin(__builtin_amdgcn_global_load_async_to_lds_b32) && \
    __has_builtin(__builtin_amdgcn_s_wait_asynccnt)
#define ATSS_HAVE_ASYNC 1
#endif
#endif
#ifndef ATSS_HAVE_ASYNC
#define ATSS_HAVE_ASYNC 0
#endif

// ---------------------------------------------------------------------------
// Kernel A: one block per batch item. Candidate generation + WMMA threshold.
// ---------------------------------------------------------------------------
__global__ void __launch_bounds__(128)
atss_candidates(const float* __restrict__ ann,
                const float* __restrict__ an0, const float* __restrict__ an1,
                const float* __restrict__ an2, const float* __restrict__ an3,
                const float* __restrict__ an4,
                int* __restrict__ wsAnch, float* __restrict__ wsIou,
                float* __restrict__ wsAx)
{
  extern __shared__ char smem[];
  float* sCiou   = (float*)smem;                 // [G_SZ][PITCH]
  float* sCax    = sCiou + G_SZ * PITCH;         // [RECS]
  float* sSums   = sCax + RECS;                  // [G_SZ]
  float* sMean   = sSums + G_SZ;                 // [G_SZ]
  float* sThresh = sMean + G_SZ;                 // [G_SZ]
  float* sAnn    = sThresh + G_SZ;               // [2*G_SZ] (l, r per gt)
  int*   sCidx   = (int*)(sAnn + 2 * G_SZ);      // [RECS]

  const int b   = blockIdx.x;
  const int tid = threadIdx.x;
  const float2* atab[NLEV] = {(const float2*)an0, (const float2*)an1,
                              (const float2*)an2, (const float2*)an3,
                              (const float2*)an4};

  if (tid < 2 * G_SZ) {
    const int g = tid >> 1, c = tid & 1;
    sAnn[tid] = ann[(size_t)b * (G_SZ * 3) + g * 3 + c];
  }
  if (tid < G_SZ) sCiou[tid * PITCH + CPG] = 0.0f;   // zero pad column
  __syncthreads();

  // Phase 1: exact top-27 per (gt, level) inside an 11-position superset.
  // Centers are a uniform grid per level, so the true top-9 positions (by the
  // exact fp32 distances the reference uses, incl. rounding of (al+ar)/2)
  // always lie inside [ctr-5, ctr+5]. Rank with (dist, index) lex order ==
  // jax.lax.top_k tie-breaking; ranks are a total order so rank<27 selects
  // exactly the reference candidate set, and rank is a unique output slot.
  for (int pair = tid; pair < G_SZ * NLEV; pair += blockDim.x) {
    const int g = pair / NLEV, level = pair % NLEV;
    const float gl = sAnn[2 * g], gr = sAnn[2 * g + 1];
    const float gx = (gl + gr) * 0.5f;
    const float s  = (float)d_stride[level];
    const int   n  = d_npos[level];
    const float fpos = gx / s - 0.5f;
    const int ctr = (int)floorf(fpos + 0.5f);
    const int lo  = min(max(ctr - 5, 0), n - 11);
    const float2* anc = atab[level];
    const int baseLocal = 3 * lo;

    float dA[33], alA[33], arA[33];
    for (int i = 0; i < 33; ++i) {
      const float2 av = anc[baseLocal + i];
      alA[i] = av.x; arA[i] = av.y;
      const float ax = (av.x + av.y) * 0.5f;
      dA[i] = fabsf(ax - gx);
    }
    for (int i = 0; i < 33; ++i) {
      const float di = dA[i];
      int rank = 0;
      for (int j = 0; j < 33; ++j)
        rank += (int)((dA[j] < di) || (dA[j] == di && j < i));
      if (rank < TOPK) {
        const float al = alA[i], ar = arA[i];
        const float ax = (al + ar) * 0.5f;
        float iw = fmaxf(fminf(ar, gr) - fmaxf(al, gl), 0.0f);
        float ua = fmaxf((ar - al) + (gr - gl) - iw, 1e-8f);
        const float iou = iw / ua;
        const int slot = level * TOPK + rank;
        const int rec  = g * CPG + slot;
        sCiou[g * PITCH + slot] = iou;
        sCax[rec] = ax;
        const bool inb = fminf(ax - gl, gr - ax) > 0.01f;
        const int ag = d_aoff[level] + baseLocal + i;
        sCidx[rec] = inb ? ag : (int)((unsigned)ag | 0x80000000u);
      }
    }
  }
  __syncthreads();

  // Phase 2: WMMA column-sum reductions. A = 16 gt-rows x 4 candidates,
  // B = ones => every D row holds that gt's running sum. Three waves, one
  // 16-gt tile each; fp32 16x16x4 keeps threshold comparisons in fp32.
  const int wv = tid >> 5, lane = tid & 31;
  if (wv < 3) {
    const int m    = wv * 16 + (lane & 15);   // gt row owned by this lane
    const int half = lane >> 4;               // K pair select per A layout
    v2f ones; ones.x = 1.0f; ones.y = 1.0f;
    v8f acc = {};
    for (int c = 0; c < 34; ++c) {
      const int k0 = c * 4 + half * 2;
      v2f a;
      a.x = sCiou[m * PITCH + k0];
      a.y = sCiou[m * PITCH + k0 + 1];        // k0+1==135 is the zero pad
      acc = __builtin_amdgcn_wmma_f32_16x16x4_f32(false, a, false, ones,
                                                  (short)0, acc, false, false);
    }
    // D layout: N=0 lives in lane 0 (rows M=0..7 in v0..7) and lane 16
    // (rows M=8..15).
    if (lane == 0) {
#pragma unroll
      for (int r = 0; r < 8; ++r) sSums[wv * 16 + r] = acc[r];
    }
    if (lane == 16) {
#pragma unroll
      for (int r = 0; r < 8; ++r) sSums[wv * 16 + 8 + r] = acc[r];
    }
  }
  __syncthreads();
  if (tid < G_SZ) sMean[tid] = sSums[tid] / 135.0f;
  __syncthreads();

  if (wv < 3) {  // second pass: sum of squared deviations (two-pass std)
    const int m    = wv * 16 + (lane & 15);
    const int half = lane >> 4;
    const float mu = sMean[m];
    v2f ones; ones.x = 1.0f; ones.y = 1.0f;
    v8f acc = {};
    for (int c = 0; c < 34; ++c) {
      const int k0 = c * 4 + half * 2;
      const float x0 = sCiou[m * PITCH + k0] - mu;
      const float x1 = sCiou[m * PITCH + k0 + 1] - mu;
      v2f a;
      a.x = x0 * x0;
      a.y = (k0 + 1 < CPG) ? x1 * x1 : 0.0f;  // mask pad: (0-mu)^2 != 0
      acc = __builtin_amdgcn_wmma_f32_16x16x4_f32(false, a, false, ones,
                                                  (short)0, acc, false, false);
    }
    if (lane == 0) {
#pragma unroll
      for (int r = 0; r < 8; ++r) sSums[wv * 16 + r] = acc[r];
    }
    if (lane == 16) {
#pragma unroll
      for (int r = 0; r < 8; ++r) sSums[wv * 16 + 8 + r] = acc[r];
    }
  }
  __syncthreads();
  if (tid < G_SZ)
    sThresh[tid] = sMean[tid] + sqrtf(sSums[tid] / 134.0f);   // std, ddof=1
  __syncthreads();

  // Phase 3: finalize records to workspace (anchor = -1 when not positive).
  const size_t recBase = (size_t)b * RECS;
  for (int idx = tid; idx < RECS; idx += blockDim.x) {
    const int g = idx / CPG, j = idx % CPG;
    const float v = sCiou[g * PITCH + j];
    const int enc = sCidx[idx];
    const bool inb = enc >= 0;
    const int ag = enc & 0x7fffffff;
    const bool pos = inb && (v >= sThresh[g]);
    wsAnch[recBase + idx] = pos ? ag : -1;
    wsIou[recBase + idx]  = v;
    wsAx[recBase + idx]   = sCax[idx];
  }
}

// ---------------------------------------------------------------------------
// Kernel B: per-batch argmax resolution + masked L1 loss. A positive record
// wins its anchor iff no other positive record for the same anchor has larger
// IoU (or equal IoU with a smaller gt index == smaller record index).
// Same anchor implies same level, so the scan is restricted to same-level
// candidate slots (48 gts x 27).
// ---------------------------------------------------------------------------
__global__ void __launch_bounds__(256)
atss_resolve(const float* __restrict__ regs, const float* __restrict__ ann,
             const int* __restrict__ wsAnch, const float* __restrict__ wsIou,
             const float* __restrict__ wsAx, float* __restrict__ wsLoss)
{
  extern __shared__ char smem[];
  int*   sA   = (int*)smem;            // [RECS]
  float* sV   = (float*)(sA + RECS);   // [RECS]
  float* sAnn = sV + RECS;             // [2*G_SZ]
  float* sRed = sAnn + 2 * G_SZ;       // [256]
  int*   sCnt = (int*)(sRed + 256);    // [256]

  const int b = blockIdx.x;
  const int tid = threadIdx.x;
  const size_t recBase = (size_t)b * RECS;

#if ATSS_HAVE_ASYNC
  // CDNA5 async copy: global -> LDS without touching VGPR data, tracked by
  // ASYNCcnt, drained with s_wait_asynccnt before the workgroup barrier.
  // Builtin signature (per hipcc diagnostic): (AS1 int*, AS3 int*, imm, imm).
  if (tid < 2 * G_SZ) {
    const int g = tid >> 1, c = tid & 1;
    const float* src = ann + (size_t)b * (G_SZ * 3) + g * 3 + c;
    typedef __attribute__((address_space(1))) int* gip_t;
    typedef __attribute__((address_space(3))) int* lip_t;
    __builtin_amdgcn_global_load_async_to_lds_b32(
        (gip_t)(uintptr_t)src, (lip_t)(uint32_t)(uintptr_t)(&sAnn[tid]), 0, 0);
  }
  __builtin_amdgcn_s_wait_asynccnt(0);
#else
  if (tid < 2 * G_SZ) {
    const int g = tid >> 1, c = tid & 1;
    sAnn[tid] = ann[(size_t)b * (G_SZ * 3) + g * 3 + c];
  }
#endif

  for (int idx = tid; idx < RECS; idx += blockDim.x) {
    const int a = wsAnch[recBase + idx];
    sA[idx] = a;
    sV[idx] = wsIou[recBase + idx];
    if (a >= 0)   // sparse gather of regressions: prefetch the lines we need
      __builtin_prefetch(&regs[((size_t)b * A_TOTAL + a) * 2], 0, 1);
  }
  __syncthreads();

  float mySum = 0.0f;
  int   myCnt = 0;
  for (int idx = tid; idx < RECS; idx += blockDim.x) {
    const int a = sA[idx];
    if (a < 0) continue;
    const float v = sV[idx];
    const int level = (idx % CPG) / TOPK;
    bool win = true;
    for (int g2 = 0; g2 < G_SZ && win; ++g2) {
      const int base = g2 * CPG + level * TOPK;
      for (int t = 0; t < TOPK; ++t) {
        const int i2 = base + t;
        if (sA[i2] != a || i2 == idx) continue;
        const float v2 = sV[i2];
        if (v2 > v || (v2 == v && i2 < idx)) { win = false; break; }
      }
    }
    if (win) {
      const int g = idx / CPG;
      const float gl = sAnn[2 * g], gr = sAnn[2 * g + 1];
      const float ax = wsAx[recBase + idx];
      const float tl = ax - gl, tr = gr - ax;
      const float* rp = regs + ((size_t)b * A_TOTAL + a) * 2;
      mySum += fabsf(rp[0] - tl) + fabsf(rp[1] - tr);
      myCnt += 1;
    }
  }

  // Deterministic tree reduction (no float atomics: replays must match).
  sRed[tid] = mySum;
  sCnt[tid] = myCnt;
  __syncthreads();
  for (int off = 128; off > 0; off >>= 1) {
    if (tid < off) {
      sRed[tid] += sRed[tid + off];
      sCnt[tid] += sCnt[tid + off];
    }
    __syncthreads();
  }
  if (tid == 0)
    wsLoss[b] = sRed[0] / (2.0f * (float)max(sCnt[0], 1));
}

// ---------------------------------------------------------------------------
// Kernel C: mean over batch, WEIGHT = 1.0.
// ---------------------------------------------------------------------------
__global__ void atss_final(const float* __restrict__ wsLoss,
                           float* __restrict__ out)
{
  if (threadIdx.x == 0 && blockIdx.x == 0)
    out[0] = 0.25f * (((wsLoss[0] + wsLoss[1]) + wsLoss[2]) + wsLoss[3]);
}

// ---------------------------------------------------------------------------
// Host launcher. Input order (setup_inputs dict): regressions, annotations,
// class_id, anchors0..anchors4. class_id is unused by the reference math.
// ---------------------------------------------------------------------------
extern "C" void kernel_launch(void* const* d_in, const int* in_sizes, int n_in,
                              void* d_out, int out_size, void* d_ws, size_t ws_size,
                              hipStream_t stream) {
  (void)in_sizes; (void)n_in; (void)out_size; (void)ws_size;
  const float* regs = (const float*)d_in[0];
  const float* ann  = (const float*)d_in[1];
  const float* an0  = (const float*)d_in[3];
  const float* an1  = (const float*)d_in[4];
  const float* an2  = (const float*)d_in[5];
  const float* an3  = (const float*)d_in[6];
  const float* an4  = (const float*)d_in[7];
  float* out = (float*)d_out;

  char* ws = (char*)d_ws;
  int*   wsAnch = (int*)ws;                                   // B*RECS ints
  float* wsIou  = (float*)(ws + (size_t)4 * B_SZ * RECS);     // B*RECS floats
  float* wsAx   = (float*)(ws + (size_t)8 * B_SZ * RECS);     // B*RECS floats
  float* wsLoss = (float*)(ws + (size_t)12 * B_SZ * RECS);    // B floats

  const size_t smemA =
      (size_t)(G_SZ * PITCH + RECS + 3 * G_SZ + 2 * G_SZ) * sizeof(float) +
      (size_t)RECS * sizeof(int);                             // ~78.9 KB
  const size_t smemB =
      (size_t)(2 * RECS + 2 * G_SZ + 256) * sizeof(float) +
      (size_t)256 * sizeof(int);                              // ~54.3 KB

  atss_candidates<<<B_SZ, 128, smemA, stream>>>(ann, an0, an1, an2, an3, an4,
                                                wsAnch, wsIou, wsAx);
  atss_resolve<<<B_SZ, 256, smemB, stream>>>(regs, ann, wsAnch, wsIou, wsAx,
                                             wsLoss);
  atss_final<<<1, 32, 0, stream>>>(wsLoss, out);
}